// LocationSenstiveAttention_26663156974074
// MI455X (gfx1250) — compile-verified
//
#include <hip/hip_runtime.h>
#include <hip/hip_bf16.h>
#include <math.h>

// ---------------- problem constants ----------------
#define BB     32
#define TT     2048
#define ENC_C  512
#define DEC_C  1024
#define HID    128
#define CONV_C 32
#define KK     31          // conv taps (padded to 32 with a zero weight row)

typedef __attribute__((ext_vector_type(2))) float v2f;
typedef __attribute__((ext_vector_type(8))) float v8f;

// LDS row stride for the 32x512 A tile: +4 floats padding -> conflict-free
// ds_load_b64 fragment reads (stride 516 floats = 2064 B; bank = 4*m distinct).
#define A_STRIDE 516

// ---------------- setup kernels ----------------
// dec_proj[b,h] = sum_k dec_state[b,k] * W_dec[k,h]          (32 x 128)
__global__ void __launch_bounds__(128)
dec_proj_kernel(const float* __restrict__ dec_state,
                const float* __restrict__ W_dec,
                float* __restrict__ dec_proj) {
    const int b = blockIdx.x;
    const int h = threadIdx.x;
    float acc = 0.f;
    const float* ds = dec_state + b * DEC_C;
    const float* wd = W_dec + h;
    #pragma unroll 8
    for (int k = 0; k < DEC_C; ++k)
        acc = fmaf(ds[k], wd[(size_t)k * HID], acc);
    dec_proj[b * HID + h] = acc;
}

// MwT[h,k] = sum_cc conv_w[cc,0,k] * W_att[cc,h]  for k<31, col 31 = 0 (128 x 32,
// stored transposed so B fragments are contiguous float2 in the K direction)
__global__ void __launch_bounds__(128)
fold_conv_kernel(const float* __restrict__ conv_w,
                 const float* __restrict__ W_att,
                 float* __restrict__ MwT) {
    const int k = blockIdx.x;     // 0..31
    const int h = threadIdx.x;    // 0..127
    float acc = 0.f;
    if (k < KK) {
        #pragma unroll
        for (int cc = 0; cc < CONV_C; ++cc)
            acc = fmaf(conv_w[cc * KK + k], W_att[cc * HID + h], acc);
    }
    MwT[h * 32 + k] = acc;
}

// WT[h,c] = W_enc[c,h]   (128 x 512) -- one-time 256 KB transpose so B fragments
// (two consecutive K values of a fixed column h) become one global_load_b64.
__global__ void __launch_bounds__(256)
transpose_wenc_kernel(const float* __restrict__ W_enc,
                      float* __restrict__ WT) {
    const int idx = blockIdx.x * 256 + threadIdx.x;   // 0 .. 65535
    const int h = idx >> 9;          // 0..127
    const int c = idx & 511;         // 0..511
    WT[h * ENC_C + c] = W_enc[(size_t)c * HID + h];
}

// ---------------- main fused energy kernel (WMMA) ----------------
// Per block: batch b, 32 consecutive t rows. 8 waves; wave w owns h-tile
// [16w,16w+16) and BOTH 16-row M-tiles, sharing each B fragment across the two
// accumulators (halves B-side global_load_b64 per wmma).
//   acc(16x16)x2 = enc_tile(32x512) @ W_enc[:,tile]  (K=512, f32 WMMA)
//                + prev_window(32x32) @ Mw[:,tile]    (K=32,  f32 WMMA)
// epilogue: x = acc + b_enc + dec_proj; e_t = sum_h tanh(x)*w_score[h] + b_score.
__global__ void __launch_bounds__(256)
energy_kernel(const float* __restrict__ enc,          // (B,T,ENC_C)
              const float* __restrict__ prevw,        // (B,T)
              const unsigned char* __restrict__ mask, // (B,T) bool
              const float* __restrict__ WT,           // (HID,ENC_C)  transposed W_enc
              const float* __restrict__ b_enc,        // (HID)
              const float* __restrict__ dec_proj,     // (B,HID)
              const float* __restrict__ MwT,          // (HID,32)     transposed folded conv
              const float* __restrict__ w_score,      // (HID)
              const float* __restrict__ b_score,      // (1)
              float* __restrict__ energy_out)         // (B,T)
{
    const int b  = blockIdx.x;
    const int t0 = blockIdx.y * 32;
    const int tid = threadIdx.x;

    __shared__ float Alds[32 * A_STRIDE];   // 32x512 enc tile (padded rows)
    __shared__ float pwlds[64];             // prev_att_w window t0-15 .. t0+48
    __shared__ float elds[32];              // per-row energy accumulators

    // cooperative load of A tile: 4096 float4 (16 per thread)
    {
        const float* src = enc + ((size_t)(b * TT + t0)) * ENC_C;
        for (int i = tid; i < 4096; i += 256) {
            const int row = i >> 7;              // 128 float4 per row
            const int c4  = (i & 127) << 2;
            const float4 v = *reinterpret_cast<const float4*>(src + (size_t)row * ENC_C + c4);
            *reinterpret_cast<float4*>(&Alds[row * A_STRIDE + c4]) = v;
        }
    }
    if (tid < 64) {
        const int t = t0 - 15 + tid;
        pwlds[tid] = (t >= 0 && t < TT) ? prevw[b * TT + t] : 0.f;
    }
    if (tid < 32) elds[tid] = 0.f;
    __syncthreads();

    const int wave  = tid >> 5;
    const int lane  = tid & 31;
    const int m     = lane & 15;   // A-fragment row
    const int n     = lane & 15;   // B/C fragment column
    const int khalf = lane >> 4;   // lanes 16-31 carry K+2 (A/B), M+8 (C)
    const int h0    = wave * 16;

    v8f acc0 = {};                 // rows t0+m
    v8f acc1 = {};                 // rows t0+16+m

    // ---- enc @ W_enc, K = 512 in steps of 4; B shared by both M-tiles ----
    const float* Ap0 = &Alds[m * A_STRIDE + 2 * khalf];
    const float* Ap1 = &Alds[(m + 16) * A_STRIDE + 2 * khalf];
    const float* Bp  = WT + (size_t)(h0 + n) * ENC_C + 2 * khalf;
    #pragma unroll 8
    for (int k0 = 0; k0 < ENC_C; k0 += 4) {
        const v2f bb = *reinterpret_cast<const v2f*>(Bp + k0);
        const v2f a0 = *reinterpret_cast<const v2f*>(Ap0 + k0);
        const v2f a1 = *reinterpret_cast<const v2f*>(Ap1 + k0);
        acc0 = __builtin_amdgcn_wmma_f32_16x16x4_f32(
            false, a0, false, bb, (short)0, acc0, false, false);
        acc1 = __builtin_amdgcn_wmma_f32_16x16x4_f32(
            false, a1, false, bb, (short)0, acc1, false, false);
    }

    // ---- prev_window @ Mw, K = 32 (tap 31 is a zero weight column) ----
    const float* Mp = MwT + (size_t)(h0 + n) * 32 + 2 * khalf;
    #pragma unroll
    for (int k0 = 0; k0 < 32; k0 += 4) {
        const int k = k0 + 2 * khalf;
        const v2f bb = *reinterpret_cast<const v2f*>(Mp + k0);
        v2f a0, a1;
        a0.x = pwlds[m + k];             // odd offsets: scalar ds loads
        a0.y = pwlds[m + k + 1];
        a1.x = pwlds[m + 16 + k];
        a1.y = pwlds[m + 16 + k + 1];
        acc0 = __builtin_amdgcn_wmma_f32_16x16x4_f32(
            false, a0, false, bb, (short)0, acc0, false, false);
        acc1 = __builtin_amdgcn_wmma_f32_16x16x4_f32(
            false, a1, false, bb, (short)0, acc1, false, false);
    }

    // ---- epilogue: tanh, score, reduce ----
    const int h   = h0 + n;
    const float wsc  = w_score[h];
    const float benc = b_enc[h];
    const float dp   = dec_proj[b * HID + h];

    #pragma unroll
    for (int r = 0; r < 8; ++r) {
        float x0 = tanhf(acc0[r] + benc + dp) * wsc;
        float x1 = tanhf(acc1[r] + benc + dp) * wsc;
        // reduce across the 16 lanes holding columns of the same row
        #pragma unroll
        for (int off = 8; off; off >>= 1) {
            x0 += __shfl_xor(x0, off, 16);
            x1 += __shfl_xor(x1, off, 16);
        }
        if (n == 0) {
            atomicAdd(&elds[r + (khalf << 3)], x0);        // ds_add_f32
            atomicAdd(&elds[16 + r + (khalf << 3)], x1);   // across 8 waves
        }
    }
    __syncthreads();

    if (tid < 32) {
        const int t = t0 + tid;
        float e = elds[tid] + b_score[0];
        if (mask[b * TT + t]) e = -INFINITY;
        energy_out[b * TT + t] = e;
    }
}

// ---------------- softmax over T per batch row ----------------
__global__ void __launch_bounds__(256)
softmax_kernel(const float* __restrict__ energy, float* __restrict__ attw) {
    const int b = blockIdx.x;
    const int tid = threadIdx.x;
    __shared__ float red[256];

    float mx = -INFINITY;
    for (int t = tid; t < TT; t += 256)
        mx = fmaxf(mx, energy[b * TT + t]);
    red[tid] = mx; __syncthreads();
    for (int s = 128; s; s >>= 1) {
        if (tid < s) red[tid] = fmaxf(red[tid], red[tid + s]);
        __syncthreads();
    }
    mx = red[0]; __syncthreads();

    float sum = 0.f;
    for (int t = tid; t < TT; t += 256) {
        const float e = __expf(energy[b * TT + t] - mx);
        attw[b * TT + t] = e;
        sum += e;
    }
    red[tid] = sum; __syncthreads();
    for (int s = 128; s; s >>= 1) {
        if (tid < s) red[tid] += red[tid + s];
        __syncthreads();
    }
    const float inv = 1.f / red[0];

    for (int t = tid; t < TT; t += 256)
        attw[b * TT + t] *= inv;
}

// ---------------- att_c = sum_t att_w * enc ----------------
__global__ void __launch_bounds__(256)
zero_attc_kernel(float* __restrict__ attc) {
    attc[blockIdx.x * 256 + threadIdx.x] = 0.f;
}

#define TSPLIT 16
__global__ void __launch_bounds__(128)
attc_kernel(const float* __restrict__ enc,
            const float* __restrict__ attw,
            float* __restrict__ attc) {
    const int b  = blockIdx.x;
    const int ts = blockIdx.y;                 // t chunk of TT/TSPLIT
    const int c  = threadIdx.x * 4;            // 128 threads x float4 = 512 cols
    const int tbeg = ts * (TT / TSPLIT);

    float4 acc = make_float4(0.f, 0.f, 0.f, 0.f);
    for (int tt = 0; tt < TT / TSPLIT; ++tt) {
        const int t = tbeg + tt;
        const float w = attw[b * TT + t];
        const float4 e = *reinterpret_cast<const float4*>(
            enc + ((size_t)(b * TT + t)) * ENC_C + c);
        acc.x = fmaf(w, e.x, acc.x);
        acc.y = fmaf(w, e.y, acc.y);
        acc.z = fmaf(w, e.z, acc.z);
        acc.w = fmaf(w, e.w, acc.w);
    }
    float* o = attc + b * ENC_C + c;
    atomicAdd(o + 0, acc.x);
    atomicAdd(o + 1, acc.y);
    atomicAdd(o + 2, acc.z);
    atomicAdd(o + 3, acc.w);
}

// ---------------- launcher ----------------
extern "C" void kernel_launch(void* const* d_in, const int* in_sizes, int n_in,
                              void* d_out, int out_size, void* d_ws, size_t ws_size,
                              hipStream_t stream) {
    const float* enc       = (const float*)d_in[0];
    const float* dec_state = (const float*)d_in[1];
    // d_in[2] = data_len: unused by the reference
    const float* prevw     = (const float*)d_in[3];
    const unsigned char* mask = (const unsigned char*)d_in[4];
    const float* W_enc     = (const float*)d_in[5];
    const float* b_enc     = (const float*)d_in[6];
    const float* W_dec     = (const float*)d_in[7];
    const float* W_att     = (const float*)d_in[8];
    const float* conv_w    = (const float*)d_in[9];
    const float* w_score   = (const float*)d_in[10];
    const float* b_score   = (const float*)d_in[11];

    float* attc = (float*)d_out;                        // (B, ENC_C)
    float* attw = (float*)d_out + BB * ENC_C;           // (B, T)

    char* ws = (char*)d_ws;
    float* dec_proj = (float*)(ws);                      // 32*128   = 16 KB
    float* MwT      = (float*)(ws + 16384);              // 128*32   = 16 KB
    float* WT       = (float*)(ws + 32768);              // 128*512  = 256 KB
    float* energy   = (float*)(ws + 32768 + 262144);     // 32*2048  = 256 KB

    dec_proj_kernel<<<dim3(BB), dim3(HID), 0, stream>>>(dec_state, W_dec, dec_proj);
    fold_conv_kernel<<<dim3(32), dim3(HID), 0, stream>>>(conv_w, W_att, MwT);
    transpose_wenc_kernel<<<dim3(ENC_C * HID / 256), dim3(256), 0, stream>>>(W_enc, WT);

    energy_kernel<<<dim3(BB, TT / 32), dim3(256), 0, stream>>>(
        enc, prevw, mask, WT, b_enc, dec_proj, MwT, w_score, b_score, energy);

    softmax_kernel<<<dim3(BB), dim3(256), 0, stream>>>(energy, attw);

    zero_attc_kernel<<<dim3(BB * ENC_C / 256), dim3(256), 0, stream>>>(attc);
    attc_kernel<<<dim3(BB, TSPLIT), dim3(128), 0, stream>>>(enc, attw, attc);
}